// GMMEnergy_80625126081176
// MI455X (gfx1250) — compile-verified
//
#include <hip/hip_runtime.h>
#include <math.h>

#define KMIX 8
#define DDIM 8
#define TRI  36
#define PHI_DIM 360          // 8 + 64 + 288
#define MROWS 32
#define ROWS 256             // MROWS * KMIX
#define FEAT 48              // 45 features padded to 48 (12 WMMA K-steps of 4)
#define PTS_PER_WAVE 16
#define WAVES_PER_BLOCK 8
#define PTS_PER_BLOCK 128

typedef __attribute__((ext_vector_type(2))) float v2f;
typedef __attribute__((ext_vector_type(8))) float v8f;

__device__ __forceinline__ int tri_idx(int i, int j) { return i * (i + 1) / 2 + j; }

// ---------------------------------------------------------------------------
// Kernel 1: per (m,k) row, fold triangular solve into quadratic-form weights.
// W[row=m*8+k][0..47] = [log_pi + lnorm - 0.5*muPmu, (P mu)_i, -mult*0.5*P_ij, 0pad]
// ---------------------------------------------------------------------------
__global__ void gmm_prep(const float* __restrict__ phi, float* __restrict__ W)
{
    int t = threadIdx.x;                 // 0..255
    int m = t >> 3, k = t & 7;
    const float* ph = phi + m * PHI_DIM;

    // log_softmax over pi_tilde, take component k
    float mx = ph[0];
    #pragma unroll
    for (int i = 1; i < KMIX; ++i) mx = fmaxf(mx, ph[i]);
    float se = 0.f;
    #pragma unroll
    for (int i = 0; i < KMIX; ++i) se += __expf(ph[i] - mx);
    float log_pi = ph[k] - (mx + __logf(se));

    float mu[DDIM];
    #pragma unroll
    for (int i = 0; i < DDIM; ++i) mu[i] = ph[KMIX + k * DDIM + i];

    float Lv[TRI];
    #pragma unroll
    for (int i = 0; i < TRI; ++i) Lv[i] = ph[KMIX + KMIX * DDIM + k * TRI + i];

    // A = inv(L) (lower triangular, packed like Lv)
    float A[TRI];
    for (int i = 0; i < DDIM; ++i) {
        float dinv = 1.0f / Lv[tri_idx(i, i)];
        A[tri_idx(i, i)] = dinv;
        for (int j = 0; j < i; ++j) {
            float s = 0.f;
            for (int p = j; p < i; ++p)
                s += Lv[tri_idx(i, p)] * A[tri_idx(p, j)];
            A[tri_idx(i, j)] = -dinv * s;
        }
    }

    // P = A^T A (symmetric, packed lower)
    float P[TRI];
    for (int i = 0; i < DDIM; ++i)
        for (int j = 0; j <= i; ++j) {
            float s = 0.f;
            for (int r = i; r < DDIM; ++r)
                s += A[tri_idx(r, i)] * A[tri_idx(r, j)];
            P[tri_idx(i, j)] = s;
        }

    // b = P mu ; c = mu^T P mu
    float b[DDIM], c = 0.f;
    for (int i = 0; i < DDIM; ++i) {
        float s = 0.f;
        for (int j = 0; j < DDIM; ++j) {
            float pij = (j <= i) ? P[tri_idx(i, j)] : P[tri_idx(j, i)];
            s += pij * mu[j];
        }
        b[i] = s;
        c += s * mu[i];
    }

    float logdet = 0.f;
    for (int i = 0; i < DDIM; ++i)
        logdet += __logf(fmaxf(fabsf(Lv[tri_idx(i, i)]), 1e-8f));
    logdet *= 2.0f;

    const float LOG2PI = 1.8378770664093453f;
    float lnorm = -0.5f * ((float)DDIM * LOG2PI + logdet);

    float* w = W + t * FEAT;
    w[0] = log_pi + lnorm - 0.5f * c;
    for (int i = 0; i < DDIM; ++i) w[1 + i] = b[i];
    for (int i = 0; i < DDIM; ++i)
        for (int j = 0; j <= i; ++j)
            w[9 + tri_idx(i, j)] = ((i == j) ? -0.5f : -1.0f) * P[tri_idx(i, j)];
    w[45] = 0.f; w[46] = 0.f; w[47] = 0.f;
}

// ---------------------------------------------------------------------------
// Kernel 2: score GEMM via V_WMMA_F32_16X16X4_F32 + fused per-lane logsumexp.
//   score(256 x Ntile) = W(256 x 48) * F(48 x Ntile)
// A-operand layout (16x4 f32): lane L -> row M=L%16, VGPR0/1 = K=(L/16)*2 +{0,1}
// B-operand layout (4x16 f32): lane L -> col N=L%16, VGPR0/1 = K=(L/16)*2 +{0,1}
// C/D layout: lane L holds cols N=L%16, rows r+8*(L/16) in 8 VGPRs
//   => each lane's 8 accumulator values are exactly the 8 k-scores of one (m,n)
// ---------------------------------------------------------------------------
__global__ void __launch_bounds__(256) gmm_score(const float* __restrict__ W,
                                                 const float* __restrict__ X,
                                                 float* __restrict__ partial,
                                                 int N)
{
    __shared__ float lds_w[ROWS * FEAT];                    // 48 KB
    __shared__ float lds_part[WAVES_PER_BLOCK * MROWS];     // 1 KB

    int tid = threadIdx.x;
    for (int i = tid; i < ROWS * FEAT; i += 256) lds_w[i] = W[i];
    __syncthreads();

    int wave = tid >> 5;
    int lane = tid & 31;
    int col  = lane & 15;      // point column within the wave's 16-point slab
    int hi   = lane >> 4;      // which half-wave
    int kb2  = hi << 1;        // K base offset {0,2} within a K=4 step

    int  n     = blockIdx.x * PTS_PER_BLOCK + wave * PTS_PER_WAVE + col;
    bool valid = (n < N);
    int  nc    = valid ? n : (N - 1);

    const float* xp = X + (size_t)nc * DDIM;
    float x[DDIM];
    #pragma unroll
    for (int i = 0; i < DDIM; ++i) x[i] = xp[i];

    // 48 features of this lane's point, kept in registers (constant indices)
    float feat[FEAT];
    feat[0] = 1.0f;
    #pragma unroll
    for (int i = 0; i < DDIM; ++i) feat[1 + i] = x[i];
    #pragma unroll
    for (int i = 0; i < DDIM; ++i)
        #pragma unroll
        for (int j = 0; j <= i; ++j)
            feat[9 + i * (i + 1) / 2 + j] = x[i] * x[j];
    feat[45] = 0.f; feat[46] = 0.f; feat[47] = 0.f;

    for (int rt = 0; rt < 16; ++rt) {      // 16 row-tiles: rows rt*16..rt*16+15
        v8f acc = {};
        const float* wrow = &lds_w[(rt * 16 + col) * FEAT];
        #pragma unroll
        for (int s = 0; s < 12; ++s) {
            v2f a;
            a.x = wrow[4 * s + kb2];
            a.y = wrow[4 * s + kb2 + 1];
            v2f bb;
            bb.x = hi ? feat[4 * s + 2] : feat[4 * s + 0];
            bb.y = hi ? feat[4 * s + 3] : feat[4 * s + 1];
            acc = __builtin_amdgcn_wmma_f32_16x16x4_f32(
                false, a, false, bb, (short)0, acc, false, false);
        }
        // per-lane logsumexp over the 8 k-scores (m = 2*rt + hi, point = col)
        float mxv = acc[0];
        #pragma unroll
        for (int r = 1; r < 8; ++r) mxv = fmaxf(mxv, acc[r]);
        float ssum = 0.f;
        #pragma unroll
        for (int r = 0; r < 8; ++r) ssum += __expf(acc[r] - mxv);
        float lm = mxv + __logf(ssum);
        lm = valid ? lm : 0.0f;
        // sum over the 16 points of each half-wave
        #pragma unroll
        for (int off = 1; off < 16; off <<= 1)
            lm += __shfl_xor(lm, off, 32);
        if (lane == 0)  lds_part[wave * MROWS + 2 * rt]     = lm;
        if (lane == 16) lds_part[wave * MROWS + 2 * rt + 1] = lm;
    }
    __syncthreads();

    if (tid < MROWS) {
        float s = 0.f;
        #pragma unroll
        for (int w = 0; w < WAVES_PER_BLOCK; ++w)
            s += lds_part[w * MROWS + tid];
        partial[blockIdx.x * MROWS + tid] = s;
    }
}

// ---------------------------------------------------------------------------
// Kernel 3: deterministic block reduction + prior + negation.
// ---------------------------------------------------------------------------
__global__ void gmm_finish(const float* __restrict__ phi,
                           const float* __restrict__ partial,
                           float* __restrict__ out, int nblocks)
{
    int m = threadIdx.x;
    if (m >= MROWS) return;
    float s = 0.f;
    for (int b = 0; b < nblocks; ++b) s += partial[b * MROWS + m];
    float pr = 0.f;
    for (int j = 0; j < PHI_DIM; ++j) {
        float v = phi[m * PHI_DIM + j];
        pr += v * v;
    }
    out[m] = -s + 0.01f * 0.5f * pr;
}

extern "C" void kernel_launch(void* const* d_in, const int* in_sizes, int n_in,
                              void* d_out, int out_size, void* d_ws, size_t ws_size,
                              hipStream_t stream)
{
    const float* phi = (const float*)d_in[0];   // (32, 360) f32
    const float* X   = (const float*)d_in[1];   // (32768, 8) f32
    float* out = (float*)d_out;                 // (32,) f32

    int N = in_sizes[1] / DDIM;
    int nblocks = (N + PTS_PER_BLOCK - 1) / PTS_PER_BLOCK;

    float* W       = (float*)d_ws;              // ROWS*FEAT floats (48 KB)
    float* partial = W + ROWS * FEAT;           // nblocks*32 floats

    gmm_prep<<<dim3(1), dim3(256), 0, stream>>>(phi, W);
    gmm_score<<<dim3(nblocks), dim3(256), 0, stream>>>(W, X, partial, N);
    gmm_finish<<<dim3(1), dim3(MROWS), 0, stream>>>(phi, partial, out, nblocks);
}